// CDDLinear_46703474377360
// MI455X (gfx1250) — compile-verified
//
#include <hip/hip_runtime.h>
#include <hip/hip_bf16.h>

typedef __attribute__((ext_vector_type(2))) float v2f;
typedef __attribute__((ext_vector_type(8))) float v8f;

#define EPSV 1e-8f
#define NS    4096
#define NCLS  500
#define NCLSP 512
#define NF    256
#define KQ    400

// workspace layout (float offsets)
#define OFF_CENT   0                          // 512*256
#define OFF_BSRC   (OFF_CENT   + NCLSP*NF)    // 512*256
#define OFF_BTGT   (OFF_BSRC   + NCLSP*NF)    // 512*256
#define OFF_CSSRC  (OFF_BTGT   + NCLSP*NF)    // 512
#define OFF_CSTGT  (OFF_CSSRC  + NCLSP)       // 512
#define OFF_MS     (OFF_CSTGT  + NCLSP)       // 512*256
#define OFF_MT     (OFF_MS     + NCLSP*NF)    // 512*256
#define OFF_NSRC   (OFF_MT     + NCLSP*NF)    // 512
#define OFF_NTGT   (OFF_NSRC   + NCLSP)       // 512
#define OFF_INTER  (OFF_NTGT   + NCLSP)       // 1024
#define OFF_INTRA  (OFF_INTER  + 1024)        // 512

__device__ __forceinline__ float wave_reduce_sum(float v) {
#pragma unroll
    for (int off = 16; off > 0; off >>= 1)
        v += __shfl_xor(v, off, 32);
    return v;
}

// 1) centroid[c,f] = sum_{k<qs[c]} queue[c,f,k] / max(qs,1)
__global__ void centroid_kernel(const float* __restrict__ queue,
                                const int* __restrict__ qsize,
                                float* __restrict__ cent) {
    int c = blockIdx.x;                  // 0..499
    int lane = threadIdx.x & 31;
    int wave = threadIdx.x >> 5;         // 0..7
    int qs = qsize[c];
    float den = fmaxf((float)qs, 1.0f);
    const float* qc = queue + (size_t)c * NF * KQ;
    for (int f = wave; f < NF; f += 8) {
        const float* qf = qc + (size_t)f * KQ;
        float s = 0.f;
        for (int k = lane; k < KQ; k += 32)
            s += (k < qs) ? qf[k] : 0.f;
        s = wave_reduce_sum(s);
        if (lane == 0) cent[c * NF + f] = s / den;
    }
}

// 2) column sums of src_y / tgt_y
__global__ void colsum_kernel(const float* __restrict__ ysrc,
                              const float* __restrict__ ytgt,
                              float* __restrict__ cs_src,
                              float* __restrict__ cs_tgt) {
    int c = blockIdx.x;                  // 0..511
    const float* y = (blockIdx.y == 0) ? ysrc : ytgt;
    float* out     = (blockIdx.y == 0) ? cs_src : cs_tgt;
    float s = 0.f;
    if (c < NCLS) {
        for (int r = threadIdx.x; r < NS; r += 256)
            s += y[(size_t)r * NCLS + c];
    }
    __shared__ float red[256];
    red[threadIdx.x] = s;
    __syncthreads();
    for (int off = 128; off > 0; off >>= 1) {
        if (threadIdx.x < off) red[threadIdx.x] += red[threadIdx.x + off];
        __syncthreads();
    }
    if (threadIdx.x == 0) out[c] = red[0];
}

// 3) C(512pad,256) = Y(4096,500)^T @ X(4096,256) via V_WMMA_F32_16X16X4_F32
//    one wave per 16x16 tile; K stepped by 4
__global__ void gemm_yx_kernel(const float* __restrict__ src_y,
                               const float* __restrict__ src_x,
                               const float* __restrict__ tgt_y,
                               const float* __restrict__ tgt_x,
                               float* __restrict__ bsrc,
                               float* __restrict__ btgt) {
    const float* Y = (blockIdx.z == 0) ? src_y : tgt_y;
    const float* X = (blockIdx.z == 0) ? src_x : tgt_x;
    float* C       = (blockIdx.z == 0) ? bsrc : btgt;
    int lane = threadIdx.x & 31;
    int wave = threadIdx.x >> 5;
    int w = blockIdx.x * 4 + wave;       // 0..511
    int mt = w >> 4, nt = w & 15;
    int c0 = mt * 16, f0 = nt * 16;
    int m = lane & 15;                   // also serves as n for B
    int ks = (lane >> 4) << 1;           // 0 (lanes 0-15) or 2 (lanes 16-31)
    int cm = c0 + m;
    float amask = (cm < NCLS) ? 1.f : 0.f;
    int cmc = (cm < NCLS) ? cm : (NCLS - 1);

    v8f acc = {};
    for (int r = 0; r < NS; r += 4) {
        int r0 = r + ks;
        v2f a, b;
        a.x = Y[(size_t)r0       * NCLS + cmc] * amask;  // A[m][ks]
        a.y = Y[(size_t)(r0 + 1) * NCLS + cmc] * amask;  // A[m][ks+1]
        b.x = X[(size_t)r0       * NF + f0 + m];         // B[ks][n]
        b.y = X[(size_t)(r0 + 1) * NF + f0 + m];         // B[ks+1][n]
        acc = __builtin_amdgcn_wmma_f32_16x16x4_f32(
                  false, a, false, b, (short)0, acc, false, false);
    }
    union { v8f v; float f[8]; } u; u.v = acc;
    int mbase = c0 + ((lane >> 4) << 3);  // lanes 16-31 hold rows M=8..15
#pragma unroll
    for (int v = 0; v < 8; ++v)
        C[(size_t)(mbase + v) * NF + f0 + m] = u.f[v];
}

// 4) mean_src / mean_tgt + per-class squared norms (zero pad rows >=500)
__global__ void combine_kernel(const float* __restrict__ cent,
                               const int* __restrict__ qsize,
                               const float* __restrict__ bsrc,
                               const float* __restrict__ btgt,
                               const float* __restrict__ cs_src,
                               const float* __restrict__ cs_tgt,
                               float* __restrict__ ms,
                               float* __restrict__ mt,
                               float* __restrict__ nsrc,
                               float* __restrict__ ntgt) {
    int c = blockIdx.x;                  // 0..511
    int f = threadIdx.x;                 // 0..255
    float vms = 0.f, vmt = 0.f;
    if (c < NCLS) {
        float num = (float)qsize[c];
        float tot = num + cs_src[c] + EPSV;
        vms = (cent[c * NF + f] * num + bsrc[(size_t)c * NF + f]) / tot;
        vmt = btgt[(size_t)c * NF + f] / (cs_tgt[c] + EPSV);
    }
    ms[(size_t)c * NF + f] = vms;
    mt[(size_t)c * NF + f] = vmt;
    __shared__ float r1[256], r2[256];
    r1[f] = vms * vms; r2[f] = vmt * vmt;
    __syncthreads();
    for (int off = 128; off > 0; off >>= 1) {
        if (f < off) { r1[f] += r1[f + off]; r2[f] += r2[f + off]; }
        __syncthreads();
    }
    if (f == 0) { nsrc[c] = r1[0]; ntgt[c] = r2[0]; }
}

// 5) intra distances: one wave per class
__global__ void intra_kernel(const float* __restrict__ ms,
                             const float* __restrict__ mt,
                             const float* __restrict__ nsrc,
                             const float* __restrict__ ntgt,
                             float* __restrict__ intra) {
    int lane = threadIdx.x & 31;
    int wave = threadIdx.x >> 5;
    int c = blockIdx.x * 8 + wave;       // 0..511
    float dot = 0.f;
#pragma unroll
    for (int j = 0; j < 8; ++j) {
        int f = lane + j * 32;
        dot += ms[(size_t)c * NF + f] * mt[(size_t)c * NF + f];
    }
    dot = wave_reduce_sum(dot);
    if (lane == 0) {
        float d = 0.f;
        if (c < NCLS) {
            float sq = nsrc[c] + ntgt[c] - 2.f * dot;
            d = sqrtf(fmaxf(sq, 1e-12f));
        }
        intra[c] = d;
    }
}

// 6) Gram tiles of mean_src @ mean_src^T via WMMA, masked sqrt-dist per tile
__global__ void inter_kernel(const float* __restrict__ ms,
                             const float* __restrict__ nsrc,
                             float* __restrict__ tile_sums) {
    int lane = threadIdx.x & 31;
    int wave = threadIdx.x >> 5;
    int w = blockIdx.x * 4 + wave;       // 0..1023
    int it = w >> 5, jt = w & 31;
    int i0 = it * 16, j0 = jt * 16;
    int m = lane & 15;
    int ks = (lane >> 4) << 1;

    v8f acc = {};
    for (int k = 0; k < NF; k += 4) {
        v2f a, b;
        a.x = ms[(size_t)(i0 + m) * NF + k + ks];      // A[m][ks]
        a.y = ms[(size_t)(i0 + m) * NF + k + ks + 1];
        b.x = ms[(size_t)(j0 + m) * NF + k + ks];      // B[ks][n] = ms[j0+n][k]
        b.y = ms[(size_t)(j0 + m) * NF + k + ks + 1];
        acc = __builtin_amdgcn_wmma_f32_16x16x4_f32(
                  false, a, false, b, (short)0, acc, false, false);
    }
    union { v8f v; float f[8]; } u; u.v = acc;
    int j = j0 + m;
    int ibase = i0 + ((lane >> 4) << 3);
    float nj = nsrc[j];
    float s = 0.f;
#pragma unroll
    for (int v = 0; v < 8; ++v) {
        int i = ibase + v;
        if (i < NCLS && j < NCLS) {
            float sq = nsrc[i] + nj - 2.f * u.f[v];
            s += sqrtf(fmaxf(sq, 1e-12f));
        }
    }
    s = wave_reduce_sum(s);
    if (lane == 0) tile_sums[w] = s;
}

// 7) deterministic final reduction -> out[0]=intra mean, out[1]=inter mean
__global__ void finalize_kernel(const float* __restrict__ intra,
                                const float* __restrict__ tile_sums,
                                float* __restrict__ out) {
    __shared__ float red[512];
    int t = threadIdx.x;                 // 0..511
    float si = intra[t];
    float sg = tile_sums[t] + tile_sums[t + 512];
    red[t] = si;
    __syncthreads();
    for (int off = 256; off > 0; off >>= 1) {
        if (t < off) red[t] += red[t + off];
        __syncthreads();
    }
    float intra_sum = red[0];
    __syncthreads();
    red[t] = sg;
    __syncthreads();
    for (int off = 256; off > 0; off >>= 1) {
        if (t < off) red[t] += red[t + off];
        __syncthreads();
    }
    if (t == 0) {
        out[0] = intra_sum / (float)NCLS;
        out[1] = red[0] / (float)(NCLS * NCLS);
    }
}

extern "C" void kernel_launch(void* const* d_in, const int* in_sizes, int n_in,
                              void* d_out, int out_size, void* d_ws, size_t ws_size,
                              hipStream_t stream) {
    const float* src_x = (const float*)d_in[0];
    const float* tgt_x = (const float*)d_in[1];
    const float* src_y = (const float*)d_in[2];
    const float* tgt_y = (const float*)d_in[3];
    const float* queue = (const float*)d_in[4];
    const int*   qsize = (const int*)d_in[5];
    float* out = (float*)d_out;
    float* ws  = (float*)d_ws;

    float* cent   = ws + OFF_CENT;
    float* bsrc   = ws + OFF_BSRC;
    float* btgt   = ws + OFF_BTGT;
    float* cssrc  = ws + OFF_CSSRC;
    float* cstgt  = ws + OFF_CSTGT;
    float* ms     = ws + OFF_MS;
    float* mt     = ws + OFF_MT;
    float* nsrc   = ws + OFF_NSRC;
    float* ntgt   = ws + OFF_NTGT;
    float* inter  = ws + OFF_INTER;
    float* intra  = ws + OFF_INTRA;

    centroid_kernel<<<NCLS, 256, 0, stream>>>(queue, qsize, cent);
    colsum_kernel<<<dim3(NCLSP, 2), 256, 0, stream>>>(src_y, tgt_y, cssrc, cstgt);
    gemm_yx_kernel<<<dim3(128, 1, 2), 128, 0, stream>>>(src_y, src_x, tgt_y, tgt_x,
                                                        bsrc, btgt);
    combine_kernel<<<NCLSP, 256, 0, stream>>>(cent, qsize, bsrc, btgt, cssrc, cstgt,
                                              ms, mt, nsrc, ntgt);
    intra_kernel<<<64, 256, 0, stream>>>(ms, mt, nsrc, ntgt, intra);
    inter_kernel<<<256, 128, 0, stream>>>(ms, nsrc, inter);
    finalize_kernel<<<1, 512, 0, stream>>>(intra, inter, out);
}